// PairWiseL2Loss_27805618274450
// MI455X (gfx1250) — compile-verified
//
#include <hip/hip_runtime.h>
#include <hip/hip_bf16.h>
#include <math.h>

// Problem constants from the reference: B=4, L=1024, n_atoms=3.
#define BATCH   4
#define SEQL    1024
#define NATOMS  3
#define NPTS    (SEQL * NATOMS)      // 3072 points per batch
#define NT      (NPTS / 16)          // 192 tiles per dimension
#define WAVES_PER_BLOCK 8            // 256 threads, wave32

typedef __attribute__((ext_vector_type(2))) float v2f;
typedef __attribute__((ext_vector_type(8))) float v8f;

#define D_CLAMP 20.0f
#define E_CLAMP 25.0f
#define EPS     1e-5f

// Workspace layout (floats):
//   w[0]          : E-sum accumulator
//   w[1..4]       : per-batch mask sums
//   w[16 + ...]   : sq_gt  [BATCH*NPTS]
//   then          : sq_pr  [BATCH*NPTS]
//   then          : m_exp  [BATCH*NPTS]
#define WS_SQG 16
#define WS_SQP (WS_SQG + BATCH * NPTS)
#define WS_MEX (WS_SQP + BATCH * NPTS)

__global__ void pwl2_init_kernel(float* __restrict__ w) {
    int t = threadIdx.x;
    if (t < 16) w[t] = 0.0f;
}

// Per-point sum-of-squares + expanded mask + per-batch mask sums.
__global__ void pwl2_precompute_kernel(const float* __restrict__ gt,
                                       const float* __restrict__ pr,
                                       const float* __restrict__ mask,
                                       float* __restrict__ w) {
    int idx = blockIdx.x * blockDim.x + threadIdx.x;   // 0 .. BATCH*NPTS-1
    int b = idx / NPTS;
    int n = idx % NPTS;

    const float* g = gt + (size_t)idx * 3;
    const float* p = pr + (size_t)idx * 3;
    float g0 = g[0], g1 = g[1], g2 = g[2];
    float p0 = p[0], p1 = p[1], p2 = p[2];
    w[WS_SQG + idx] = g0 * g0 + g1 * g1 + g2 * g2;
    w[WS_SQP + idx] = p0 * p0 + p1 * p1 + p2 * p2;

    float m = mask[b * SEQL + n / NATOMS];
    w[WS_MEX + idx] = m;

    // Wave-reduce mask sum (NPTS % 32 == 0 -> whole wave is one batch).
    float ms = m;
    #pragma unroll
    for (int off = 16; off > 0; off >>= 1)
        ms += __shfl_xor(ms, off, 32);
    if ((threadIdx.x & 31) == 0)
        atomicAdd(&w[1 + b], ms);
}

// One wave per 16x16 tile of the N x N pair matrix; Gram tiles via
// V_WMMA_F32_16X16X4_F32 (K=3 zero-padded to 4).
__global__ void pwl2_tiles_kernel(const float* __restrict__ gt,
                                  const float* __restrict__ pr,
                                  const float* __restrict__ w) {
    __shared__ float wsum[WAVES_PER_BLOCK];

    int wave = (blockIdx.x * blockDim.x + threadIdx.x) >> 5;
    int wid  = threadIdx.x >> 5;
    int lane = threadIdx.x & 31;

    int b   = wave / (NT * NT);
    int rem = wave % (NT * NT);
    int i0  = (rem / NT) * 16;
    int j0  = (rem % NT) * 16;

    int half = lane >> 4;      // 0: K=0,1   1: K=2,pad
    int mn   = lane & 15;

    size_t baseP = (size_t)b * NPTS;

    // ---- A fragment (rows i0..i0+15), ISA 16x4 f32 layout.
    //      Load all 3 coords unconditionally (one clause, no EXEC games),
    //      then select per K-half. ----
    const float* ap = pr + (baseP + i0 + mn) * 3;
    const float* ag = gt + (baseP + i0 + mn) * 3;
    float ap0 = ap[0], ap1 = ap[1], ap2 = ap[2];
    float ag0 = ag[0], ag1 = ag[1], ag2 = ag[2];
    v2f Ap, Ag;
    Ap.x = half ? ap2  : ap0;
    Ap.y = half ? 0.0f : ap1;
    Ag.x = half ? ag2  : ag0;
    Ag.y = half ? 0.0f : ag1;

    // ---- B fragment (cols j0..j0+15), 4x16 f32 layout (mirror of A) ----
    const float* bp = pr + (baseP + j0 + mn) * 3;
    const float* bg = gt + (baseP + j0 + mn) * 3;
    float bp0 = bp[0], bp1 = bp[1], bp2 = bp[2];
    float bg0 = bg[0], bg1 = bg[1], bg2 = bg[2];
    v2f Bp, Bg;
    Bp.x = half ? bp2  : bp0;
    Bp.y = half ? 0.0f : bp1;
    Bg.x = half ? bg2  : bg0;
    Bg.y = half ? 0.0f : bg1;

    // ---- Gram tiles: D = A x B + 0 ----
    v8f Cp = {};
    v8f Cg = {};
    Cp = __builtin_amdgcn_wmma_f32_16x16x4_f32(false, Ap, false, Bp,
                                               (short)0, Cp, false, false);
    Cg = __builtin_amdgcn_wmma_f32_16x16x4_f32(false, Ag, false, Bg,
                                               (short)0, Cg, false, false);

    // ---- Elementwise tail: this lane owns column j = j0+mn, rows
    //      i0 + 8*half + v for v = 0..7 (C/D layout). Raw v_sqrt_f32
    //      (~1 ulp) -- skip the IEEE fixup expansion. ----
    int j = j0 + mn;
    float sqp_j = w[WS_SQP + baseP + j];
    float sqg_j = w[WS_SQG + baseP + j];
    float m_j   = w[WS_MEX + baseP + j];

    int rbase = i0 + half * 8;
    float s = 0.0f;
    #pragma unroll
    for (int v = 0; v < 8; ++v) {
        size_t i = baseP + rbase + v;
        float d2p = w[WS_SQP + i] + sqp_j - 2.0f * Cp[v];
        float d2g = w[WS_SQG + i] + sqg_j - 2.0f * Cg[v];
        d2p = fmaxf(d2p, 0.0f);
        d2g = fmaxf(d2g, 0.0f);
        float dp = fminf(__builtin_amdgcn_sqrtf(d2p), D_CLAMP);
        float dg = __builtin_amdgcn_sqrtf(d2g);
        float df = dp - dg;
        float e  = fminf(df * df, E_CLAMP);
        s += e * w[WS_MEX + i] * m_j;
    }

    // Wave reduction, then block-stage through LDS: one atomic per block.
    #pragma unroll
    for (int off = 16; off > 0; off >>= 1)
        s += __shfl_xor(s, off, 32);
    if (lane == 0) wsum[wid] = s;
    __syncthreads();
    if (wid == 0) {
        float t = (lane < WAVES_PER_BLOCK) ? wsum[lane] : 0.0f;
        #pragma unroll
        for (int off = WAVES_PER_BLOCK / 2; off > 0; off >>= 1)
            t += __shfl_xor(t, off, 32);
        if (lane == 0)
            atomicAdd((float*)&w[0], t);
    }
}

__global__ void pwl2_finalize_kernel(const float* __restrict__ w,
                                     float* __restrict__ out) {
    if (threadIdx.x == 0 && blockIdx.x == 0) {
        float pm = 0.0f;
        #pragma unroll
        for (int b = 0; b < BATCH; ++b)
            pm += w[1 + b] * w[1 + b];
        out[0] = w[0] / (pm + EPS);
    }
}

extern "C" void kernel_launch(void* const* d_in, const int* in_sizes, int n_in,
                              void* d_out, int out_size, void* d_ws, size_t ws_size,
                              hipStream_t stream) {
    const float* gt   = (const float*)d_in[0];   // ground_truth (B,L,3,3)
    const float* pr   = (const float*)d_in[1];   // predictions  (B,L,3,3)
    const float* mask = (const float*)d_in[2];   // mask (B,L)
    float* out = (float*)d_out;
    float* w   = (float*)d_ws;

    pwl2_init_kernel<<<1, 32, 0, stream>>>(w);

    int nPoints = BATCH * NPTS;                      // 12288
    pwl2_precompute_kernel<<<nPoints / 256, 256, 0, stream>>>(gt, pr, mask, w);

    int nWaves  = BATCH * NT * NT;                   // 147456
    int nBlocks = nWaves / WAVES_PER_BLOCK;          // 18432
    pwl2_tiles_kernel<<<nBlocks, 32 * WAVES_PER_BLOCK, 0, stream>>>(gt, pr, w);

    pwl2_finalize_kernel<<<1, 32, 0, stream>>>(w, out);
}